// MoEBlock_75127567941882
// MI455X (gfx1250) — compile-verified
//
#include <hip/hip_runtime.h>
#include <stdint.h>

typedef __attribute__((ext_vector_type(2))) float v2f;
typedef __attribute__((ext_vector_type(8))) float v8f;

#define T_TOK 4096   // B*S tokens
#define DDIM  1024
#define FDIM  4096
#define NEXP  4
#define RLOW  4

// ---------------------------------------------------------------------------
// Kernel 1: router (softmax + top-2 combine weights) + lora_low = x @ A^T
// ---------------------------------------------------------------------------
__global__ __launch_bounds__(256) void moe_router(
    const float* __restrict__ x, const float* __restrict__ Wg,
    const float* __restrict__ bg, const float* __restrict__ A,
    float* __restrict__ w_ws, float* __restrict__ ll_ws)
{
  __shared__ float xs[DDIM];
  __shared__ float vals[32];
  const int t   = blockIdx.x;
  const int tid = threadIdx.x;

  for (int i = tid; i < DDIM; i += 256) xs[i] = x[(size_t)t * DDIM + i];
  __syncthreads();

  const int wave = tid >> 5;
  const int lane = tid & 31;
  for (int row = wave; row < 4 + NEXP * RLOW; row += 8) {
    const float* wr = (row < 4) ? (Wg + (size_t)row * DDIM)
                                : (A + (size_t)(row - 4) * DDIM);
    float acc = 0.f;
    for (int d = lane; d < DDIM; d += 32) acc += xs[d] * wr[d];
    for (int off = 16; off > 0; off >>= 1) acc += __shfl_down(acc, off, 32);
    if (lane == 0) vals[row] = acc;
  }
  __syncthreads();

  if (tid < NEXP * RLOW) ll_ws[(size_t)t * 16 + tid] = vals[4 + tid];
  if (tid == 0) {
    float l0 = vals[0] + bg[0], l1 = vals[1] + bg[1];
    float l2 = vals[2] + bg[2], l3 = vals[3] + bg[3];
    float mx = fmaxf(fmaxf(l0, l1), fmaxf(l2, l3));
    float p[4] = {__expf(l0 - mx), __expf(l1 - mx), __expf(l2 - mx), __expf(l3 - mx)};
    float s = p[0] + p[1] + p[2] + p[3];
    int i1 = 0;
    for (int e = 1; e < 4; ++e) if (p[e] > p[i1]) i1 = e;
    int i2 = -1;
    for (int e = 0; e < 4; ++e) if (e != i1 && (i2 < 0 || p[e] > p[i2])) i2 = e;
    for (int e = 0; e < 4; ++e)
      w_ws[(size_t)t * 4 + e] = (e == i1 || e == i2) ? p[e] / s : 0.f;
  }
}

// ---------------------------------------------------------------------------
// Kernel 2/3: fp32 WMMA GEMM  Out[M][N] = Amat[M][K] * Bmat[N][K]^T
//  - async Global->LDS staging (global_load_async_to_lds_b128, ASYNCcnt),
//    double-buffered, prefetch of tile i+1 overlaps WMMA of tile i
//  - LDS layout [BK/4][rows][4]: one ds_load_b64 per WMMA fragment,
//    conflict-free (lanes 0-15 banks {4m,4m+1}, lanes 16-31 {4m+2,4m+3})
//  - FUSE_G: epilogue g = sum_e w*relu(base + lora)
// ---------------------------------------------------------------------------
template <bool FUSE_G>
__global__ __launch_bounds__(256) void moe_gemm(
    const float* __restrict__ Amat,   // [M][K]
    const float* __restrict__ Bmat,   // [N][K]
    float* __restrict__ Out,          // [M][N_total]
    int K, int N_total,
    const float* __restrict__ w_ws,   // [T][4]
    const float* __restrict__ ll_ws,  // [T][16]
    const float* __restrict__ LoraB)  // [E][F][R]
{
  constexpr int BM = 128, BN = 64, BK = 64;
  constexpr int KG = BK / 4;                        // 16 K-groups of 4
  __shared__ __align__(16) float sX[2][KG * BM * 4]; // 2 x 32KB
  __shared__ __align__(16) float sW[2][KG * BN * 4]; // 2 x 16KB
  __shared__ float sw[BM * NEXP];
  __shared__ float sll[BM * NEXP * RLOW];
  __shared__ float sB[BN * NEXP * RLOW];

  const int tid   = threadIdx.x;
  const int nb    = blockIdx.x * BN;
  const int tb    = blockIdx.y * BM;
  const int wave  = tid >> 5;
  const int lane  = tid & 31;
  const int hi    = lane >> 4;
  const int l16   = lane & 15;
  const int waveM = (wave & 3) * 32;   // 4 waves tile M
  const int waveN = (wave >> 2) * 32;  // 2 waves tile N

  // LDS byte addresses (low 32 bits of flat address == LDS offset)
  const unsigned ldsX = (unsigned)(uintptr_t)&sX[0][0];
  const unsigned ldsW = (unsigned)(uintptr_t)&sW[0][0];

  if constexpr (FUSE_G) {
    for (int i = tid; i < BM * NEXP; i += 256)
      sw[i] = w_ws[(size_t)tb * NEXP + i];
    for (int i = tid; i < BM * NEXP * RLOW; i += 256)
      sll[i] = ll_ws[(size_t)tb * (NEXP * RLOW) + i];
    for (int i = tid; i < BN * NEXP * RLOW; i += 256) {
      int n = i >> 4, er = i & 15, e = er >> 2, r = er & 3;
      sB[i] = LoraB[((size_t)e * FDIM + (nb + n)) * RLOW + r];
    }
  }

  // Issue async Global->LDS copies for one BKx(BM|BN) tile pair into buffer b.
  auto prefetch = [&](int k0, int b) {
    const float* baseA = Amat + (size_t)tb * K + k0;
    const float* baseB = Bmat + (size_t)nb * K + k0;
#pragma unroll
    for (int it = 0; it < (KG * BM) / 256; ++it) {          // 8 x 4KB
      int lin = tid + 256 * it;
      int kk4 = lin & (KG - 1);
      int m   = lin >> 4;
      unsigned voff = (unsigned)((m * K + kk4 * 4) * 4);
      unsigned dst  = ldsX + (unsigned)(b * (KG * BM * 16) + (kk4 * BM + m) * 16);
      asm volatile("global_load_async_to_lds_b128 %0, %1, %2"
                   :: "v"(dst), "v"(voff), "s"(baseA) : "memory");
    }
#pragma unroll
    for (int it = 0; it < (KG * BN) / 256; ++it) {          // 4 x 4KB
      int lin = tid + 256 * it;
      int kk4 = lin & (KG - 1);
      int n   = lin >> 4;
      unsigned voff = (unsigned)((n * K + kk4 * 4) * 4);
      unsigned dst  = ldsW + (unsigned)(b * (KG * BN * 16) + (kk4 * BN + n) * 16);
      asm volatile("global_load_async_to_lds_b128 %0, %1, %2"
                   :: "v"(dst), "v"(voff), "s"(baseB) : "memory");
    }
  };

  v8f acc[2][2] = {};
  const int nIter = K / BK;

  prefetch(0, 0);
  for (int i = 0; i < nIter; ++i) {
    const int b = i & 1;
    asm volatile("s_wait_asynccnt 0x0" ::: "memory");  // buffer b landed
    __syncthreads();                                   // visible to all waves
    if (i + 1 < nIter) prefetch((i + 1) * BK, (i + 1) & 1);  // overlaps WMMA

#pragma unroll
    for (int kg = 0; kg < KG; ++kg) {
      v2f a[2], bb[2];
#pragma unroll
      for (int q = 0; q < 2; ++q) {
        const int m = waveM + q * 16 + l16;
        a[q]  = *(const v2f*)&sX[b][(kg * BM + m) * 4 + 2 * hi];
        const int n = waveN + q * 16 + l16;
        bb[q] = *(const v2f*)&sW[b][(kg * BN + n) * 4 + 2 * hi];
      }
#pragma unroll
      for (int q = 0; q < 2; ++q)
#pragma unroll
        for (int j = 0; j < 2; ++j)
          acc[q][j] = __builtin_amdgcn_wmma_f32_16x16x4_f32(
              false, a[q], false, bb[j], (short)0, acc[q][j], false, false);
    }
  }

  // Epilogue: C layout => VGPR v, lane L: row = v + 8*(L/16), col = L%16.
#pragma unroll
  for (int q = 0; q < 2; ++q) {
#pragma unroll
    for (int j = 0; j < 2; ++j) {
#pragma unroll
      for (int v = 0; v < 8; ++v) {
        const int m = waveM + q * 16 + v + 8 * hi;
        const int n = waveN + j * 16 + l16;
        float val = acc[q][j][v];
        if constexpr (FUSE_G) {
          float g = 0.f;
#pragma unroll
          for (int e = 0; e < NEXP; ++e) {
            float lora = 0.f;
#pragma unroll
            for (int r = 0; r < RLOW; ++r)
              lora += sll[m * 16 + e * 4 + r] * sB[n * 16 + e * 4 + r];
            float h = val + lora;
            h = h > 0.f ? h : 0.f;
            g += sw[m * 4 + e] * h;
          }
          val = g;
        }
        Out[(size_t)(tb + m) * N_total + (nb + n)] = val;
      }
    }
  }
}

// ---------------------------------------------------------------------------
extern "C" void kernel_launch(void* const* d_in, const int* in_sizes, int n_in,
                              void* d_out, int out_size, void* d_ws, size_t ws_size,
                              hipStream_t stream) {
  (void)in_sizes; (void)n_in; (void)out_size; (void)ws_size;
  const float* x  = (const float*)d_in[0];
  const float* Wg = (const float*)d_in[1];
  const float* bg = (const float*)d_in[2];
  const float* Wi = (const float*)d_in[3];
  const float* Wo = (const float*)d_in[4];
  const float* A  = (const float*)d_in[5];
  const float* Bm = (const float*)d_in[6];
  float* out = (float*)d_out;

  float* ws    = (float*)d_ws;
  float* w_ws  = ws;                                  // [T][4]
  float* ll_ws = w_ws + (size_t)T_TOK * NEXP;         // [T][16]
  float* g_ws  = ll_ws + (size_t)T_TOK * NEXP * RLOW; // [T][F]

  moe_router<<<T_TOK, 256, 0, stream>>>(x, Wg, bg, A, w_ws, ll_ws);

  dim3 g2(FDIM / 64, T_TOK / 128);
  moe_gemm<true><<<g2, 256, 0, stream>>>(x, Wi, g_ws, DDIM, FDIM,
                                         w_ws, ll_ws, Bm);

  dim3 g3(DDIM / 64, T_TOK / 128);
  moe_gemm<false><<<g3, 256, 0, stream>>>(g_ws, Wo, out, FDIM, DDIM,
                                          nullptr, nullptr, nullptr);
}